// SelfAttentionAggregationLayer_88545045775034
// MI455X (gfx1250) — compile-verified
//
#include <hip/hip_runtime.h>
#include <math.h>

// Problem constants (from reference): B=4, S=512, D=128, H=8, HD=16
#define BB 4
#define SS 512
#define DD 128
#define HH 8
#define HD 16

// LDS stage row stride for st_emb tiles: 132 floats = 528 B.
//  - 528 % 16 == 0  -> b128 async writes stay 16B aligned
//  - 132 % 64 == 4  -> 16 A-fragment rows map to 16 distinct LDS banks
#define STG 132

typedef float v2f __attribute__((ext_vector_type(2)));
typedef float v8f __attribute__((ext_vector_type(8)));

// V_WMMA_F32_16X16X4_F32:  D(16x16 f32) = A(16x4 f32) * B(4x16 f32) + C
__device__ __forceinline__ v8f wmma4(v2f a, v2f b, v8f c) {
    return __builtin_amdgcn_wmma_f32_16x16x4_f32(false, a, false, b, (short)0, c,
                                                 false, false);
}

__device__ __forceinline__ v8f vzero8() {
    v8f z = {0.f, 0.f, 0.f, 0.f, 0.f, 0.f, 0.f, 0.f};
    return z;
}

// Async copy 16 contiguous bytes/lane from global to LDS (ASYNCcnt-tracked).
__device__ __forceinline__ void async_b128(unsigned lds_off, const float* gptr) {
    asm volatile("global_load_async_to_lds_b128 %0, %1, off th:TH_LOAD_NT"
                 :: "v"(lds_off), "v"(gptr) : "memory");
}
__device__ __forceinline__ void wait_async_le16() {
    asm volatile("s_wait_asynccnt 0x10" ::: "memory");
}
__device__ __forceinline__ void wait_async_0() {
    asm volatile("s_wait_asynccnt 0x0" ::: "memory");
}

// ---------------------------------------------------------------------------
// K1: Q/K/V projection.  x[2048,128] @ w[128,128] (+bias), one 16x16 tile per
// wave; n-tile == head (HD==16).  Output layout [B,H,S,HD].
// ---------------------------------------------------------------------------
__global__ __launch_bounds__(32) void qkv_kernel(
    const float* __restrict__ x,
    const float* __restrict__ wq, const float* __restrict__ bq,
    const float* __restrict__ wk, const float* __restrict__ bk,
    const float* __restrict__ wv, const float* __restrict__ bv,
    float* __restrict__ Q, float* __restrict__ K, float* __restrict__ V) {
    const int lane = threadIdx.x & 31;
    const int n    = lane & 15;
    const int half = lane >> 4;
    const int q0   = blockIdx.x * 16; // token tile over B*S
    const int h    = blockIdx.y;      // head == n-tile
    const int n0   = h * HD;

    v8f cq = vzero8(), ck = vzero8(), cv = vzero8();
    for (int kb = 0; kb < DD; kb += 4) {
        const int ka = kb + 2 * half;
        v2f a = *(const v2f*)(x + (q0 + n) * DD + ka);
        v2f bwq, bwk, bwv;
        bwq.x = wq[(ka    ) * DD + n0 + n];
        bwq.y = wq[(ka + 1) * DD + n0 + n];
        bwk.x = wk[(ka    ) * DD + n0 + n];
        bwk.y = wk[(ka + 1) * DD + n0 + n];
        bwv.x = wv[(ka    ) * DD + n0 + n];
        bwv.y = wv[(ka + 1) * DD + n0 + n];
        cq = wmma4(a, bwq, cq);
        ck = wmma4(a, bwk, ck);
        cv = wmma4(a, bwv, cv);
    }
    const float aq = bq[n0 + n], ak = bk[n0 + n], av = bv[n0 + n];
    for (int r = 0; r < 8; ++r) {
        const int t = q0 + r + 8 * half;
        const int b = t >> 9, s = t & (SS - 1);
        const int idx = ((b * HH + h) * SS + s) * HD + n;
        Q[idx] = cq[r] + aq;
        K[idx] = ck[r] + ak;
        V[idx] = cv[r] + av;
    }
}

// ---------------------------------------------------------------------------
// K2: scores[b,h,q,k] = (Q.K^T)/4 + st_emb[b,q,k,:].wst[:,h], masked.
// One wave per (b, 16q, 16k) tile.  st_emb streamed once via double-buffered
// global_load_async_to_lds_b128 (NT, coalesced 512B rows), WMMA A-fragments
// read back from bank-conflict-free padded LDS rows.
// ---------------------------------------------------------------------------
__global__ __launch_bounds__(32) void scores_kernel(
    const float* __restrict__ st_emb, const float* __restrict__ wst,
    const unsigned char* __restrict__ mask,
    const float* __restrict__ Q, const float* __restrict__ K,
    float* __restrict__ scores) {
    __shared__ float stage[2][16 * STG];       // 2 x 8.25 KB double buffer
    __shared__ float bias_lds[16 * 16 * HH];   // [q_local][k_local][h], 8 KB
    const int lane = threadIdx.x & 31;
    const int n    = lane & 15;
    const int half = lane >> 4;
    const int k0 = blockIdx.x * 16;
    const int q0 = blockIdx.y * 16;
    const int b  = blockIdx.z;

    const unsigned stage_addr = (unsigned)(size_t)&stage[0][0];

    // Preload wst as B-fragments (heads padded 8 -> 16 with zeros); resident.
    v2f wstB[32];
    for (int i = 0; i < 32; ++i) {
        const int ka = i * 4 + 2 * half;
        wstB[i].x = (n < HH) ? wst[(ka    ) * HH + n] : 0.f;
        wstB[i].y = (n < HH) ? wst[(ka + 1) * HH + n] : 0.f;
    }

    // Issue the async copy of one 16x128 f32 tile (16 rows, one b128/lane/row).
    auto issue = [&](int ql, int sel) {
        const float* gbase =
            st_emb + (((size_t)b * SS + (q0 + ql)) * SS + k0) * DD + lane * 4;
        unsigned lbase = stage_addr + (unsigned)sel * (16 * STG * 4) + lane * 16;
#pragma unroll
        for (int kl = 0; kl < 16; ++kl)
            async_b128(lbase + kl * (STG * 4), gbase + kl * DD);
    };

    // Phase 1: per-head spatio-temporal bias, double-buffered stream.
    issue(0, 0);
    for (int ql = 0; ql < 16; ++ql) {
        if (ql + 1 < 16) {
            issue(ql + 1, (ql + 1) & 1);
            wait_async_le16();   // current tile landed; prefetch in flight
        } else {
            wait_async_0();
        }
        const float* sb = &stage[ql & 1][0];
        v8f cb = vzero8();
#pragma unroll
        for (int i = 0; i < 32; ++i) {
            v2f a = *(const v2f*)(sb + n * STG + i * 4 + 2 * half);
            cb = wmma4(a, wstB[i], cb);   // C rows = k_local, lanes = head
        }
        if (n < HH) {
            for (int r = 0; r < 8; ++r)
                bias_lds[((ql * 16) + r + 8 * half) * HH + n] = cb[r];
        }
    }
    __syncthreads();

    // Phase 2: Q.K^T per head, combine with bias, mask, write scores (L2).
    const unsigned char mk = mask[b * SS + k0 + n];
    const float scl = 0.25f;                          // 1/sqrt(HD)
    for (int h = 0; h < HH; ++h) {
        v8f cs = vzero8();
        const float* qb = Q + ((b * HH + h) * SS + q0 + n) * HD;
        const float* kb = K + ((b * HH + h) * SS + k0 + n) * HD;
        for (int db = 0; db < HD; db += 4) {
            const int ka = db + 2 * half;
            v2f a  = *(const v2f*)(qb + ka);   // A[m=q][d]
            v2f bt = *(const v2f*)(kb + ka);   // B[d][n=k] = K[k][d]
            cs = wmma4(a, bt, cs);
        }
        for (int r = 0; r < 8; ++r) {
            const int m = r + 8 * half;        // q_local
            float sc = cs[r] * scl + bias_lds[(m * 16 + n) * HH + h];
            if (mk) sc = -10000.f;
            scores[(size_t)((b * HH + h) * SS + q0 + m) * SS + k0 + n] = sc;
        }
    }
}

// ---------------------------------------------------------------------------
// K3: softmax over k (512) per (b,h,q) row, in place.  Wave per row.
// ---------------------------------------------------------------------------
__global__ __launch_bounds__(256) void softmax_kernel(float* __restrict__ scores) {
    const int lane = threadIdx.x & 31;
    const int w    = threadIdx.x >> 5;
    const int row  = blockIdx.x * 8 + w;      // B*H*S rows
    float* p = scores + (size_t)row * SS;

    float4 v[4];
    float m = -3.0e38f;
    for (int j = 0; j < 4; ++j) {
        v[j] = *(const float4*)(p + (lane + 32 * j) * 4);
        m = fmaxf(m, fmaxf(fmaxf(v[j].x, v[j].y), fmaxf(v[j].z, v[j].w)));
    }
    for (int off = 16; off; off >>= 1) m = fmaxf(m, __shfl_xor(m, off, 32));
    float s = 0.f;
    for (int j = 0; j < 4; ++j) {
        v[j].x = __expf(v[j].x - m); v[j].y = __expf(v[j].y - m);
        v[j].z = __expf(v[j].z - m); v[j].w = __expf(v[j].w - m);
        s += v[j].x + v[j].y + v[j].z + v[j].w;
    }
    for (int off = 16; off; off >>= 1) s += __shfl_xor(s, off, 32);
    const float inv = 1.f / s;
    for (int j = 0; j < 4; ++j) {
        float4 o = v[j];
        o.x *= inv; o.y *= inv; o.z *= inv; o.w *= inv;
        *(float4*)(p + (lane + 32 * j) * 4) = o;
    }
}

// ---------------------------------------------------------------------------
// K4: out = attn @ V, per (b,h,16q) tile; V[b,h] (32 KB) async-staged to LDS
// once per block, then 128 WMMAs over the 512 keys read B-frags from LDS.
// ---------------------------------------------------------------------------
__global__ __launch_bounds__(32) void av_kernel(
    const float* __restrict__ attn, const float* __restrict__ V,
    float* __restrict__ attn_out) {
    __shared__ float vtile[SS * HD];           // 32 KB
    const int lane = threadIdx.x & 31;
    const int n    = lane & 15;
    const int half = lane >> 4;
    const int q0 = blockIdx.x * 16;  // within S
    const int h  = blockIdx.y;
    const int b  = blockIdx.z;

    // Stage V[b,h][512][16] into LDS: 64 coalesced async b128 copies.
    {
        const unsigned vbase = (unsigned)(size_t)&vtile[0] + lane * 16;
        const float* gp = V + (size_t)((b * HH + h) * SS) * HD + lane * 4;
#pragma unroll 8
        for (int it = 0; it < 64; ++it)
            async_b128(vbase + it * 512, gp + it * 128);
        wait_async_0();
    }

    const float* arow = attn + (size_t)((b * HH + h) * SS + q0) * SS;
    v8f c = vzero8();
    for (int kb = 0; kb < SS; kb += 4) {
        const int ka = kb + 2 * half;
        v2f a = *(const v2f*)(arow + (size_t)n * SS + ka);  // A[q][key]
        v2f bt;                                             // B[key][d]
        bt.x = vtile[(ka    ) * HD + n];
        bt.y = vtile[(ka + 1) * HD + n];
        c = wmma4(a, bt, c);
    }
    for (int r = 0; r < 8; ++r) {
        const int m = r + 8 * half;
        attn_out[(size_t)(b * SS + q0 + m) * DD + h * HD + n] = c[r];
    }
}

// ---------------------------------------------------------------------------
// K5: y = attn_out @ wo + bo + x (residual).
// ---------------------------------------------------------------------------
__global__ __launch_bounds__(32) void oproj_kernel(
    const float* __restrict__ attn_out, const float* __restrict__ wo,
    const float* __restrict__ bo, const float* __restrict__ x,
    float* __restrict__ y) {
    const int lane = threadIdx.x & 31;
    const int n    = lane & 15;
    const int half = lane >> 4;
    const int q0 = blockIdx.x * 16;       // over B*S
    const int n0 = blockIdx.y * 16;

    v8f c = vzero8();
    for (int kb = 0; kb < DD; kb += 4) {
        const int ka = kb + 2 * half;
        v2f a = *(const v2f*)(attn_out + (q0 + n) * DD + ka);
        v2f bw;
        bw.x = wo[(ka    ) * DD + n0 + n];
        bw.y = wo[(ka + 1) * DD + n0 + n];
        c = wmma4(a, bw, c);
    }
    const float add = bo[n0 + n];
    for (int r = 0; r < 8; ++r) {
        const int row = q0 + r + 8 * half;
        y[row * DD + n0 + n] = c[r] + add + x[row * DD + n0 + n];
    }
}

// ---------------------------------------------------------------------------
// K6: layernorm over D=128 per row; wave per row, 4 elems per lane.
// ---------------------------------------------------------------------------
__global__ __launch_bounds__(256) void ln_kernel(
    const float* __restrict__ y, const float* __restrict__ g,
    const float* __restrict__ be, float* __restrict__ out) {
    const int lane = threadIdx.x & 31;
    const int w    = threadIdx.x >> 5;
    const int row  = blockIdx.x * 8 + w;   // B*S rows
    const float* p = y + row * DD;

    float4 v = *(const float4*)(p + lane * 4);
    float s = v.x + v.y + v.z + v.w;
    for (int off = 16; off; off >>= 1) s += __shfl_xor(s, off, 32);
    const float mu = s * (1.f / DD);
    float4 d; d.x = v.x - mu; d.y = v.y - mu; d.z = v.z - mu; d.w = v.w - mu;
    float vv = d.x * d.x + d.y * d.y + d.z * d.z + d.w * d.w;
    for (int off = 16; off; off >>= 1) vv += __shfl_xor(vv, off, 32);
    const float rs = rsqrtf(vv * (1.f / DD) + 1e-5f);
    float4 gg = *(const float4*)(g + lane * 4);
    float4 bb = *(const float4*)(be + lane * 4);
    float4 o;
    o.x = d.x * rs * gg.x + bb.x;
    o.y = d.y * rs * gg.y + bb.y;
    o.z = d.z * rs * gg.z + bb.z;
    o.w = d.w * rs * gg.w + bb.w;
    *(float4*)(out + row * DD + lane * 4) = o;
}

// ---------------------------------------------------------------------------
extern "C" void kernel_launch(void* const* d_in, const int* in_sizes, int n_in,
                              void* d_out, int out_size, void* d_ws, size_t ws_size,
                              hipStream_t stream) {
    const float* x      = (const float*)d_in[0];
    const float* st_emb = (const float*)d_in[1];
    const unsigned char* mask = (const unsigned char*)d_in[2];  // jax bool, 1B
    const float* wq  = (const float*)d_in[3];
    const float* bq  = (const float*)d_in[4];
    const float* wk  = (const float*)d_in[5];
    const float* bk  = (const float*)d_in[6];
    const float* wv  = (const float*)d_in[7];
    const float* bv  = (const float*)d_in[8];
    const float* wo  = (const float*)d_in[9];
    const float* bo  = (const float*)d_in[10];
    const float* wst = (const float*)d_in[11];
    const float* lng = (const float*)d_in[12];
    const float* lnb = (const float*)d_in[13];

    // Workspace (floats): Q,K,V,attn_out,y = 5*262144; scores = 8388608
    // total ~38.8 MB.  scores (33.5 MB) stays resident in the 192 MB L2;
    // the 536 MB st_emb stream is NT-hinted so it does not evict them.
    float* ws = (float*)d_ws;
    float* Q        = ws;
    float* K        = ws + 262144;
    float* V        = ws + 524288;
    float* attn_out = ws + 786432;
    float* y        = ws + 1048576;
    float* scores   = ws + 1310720;

    qkv_kernel<<<dim3(128, 8), 32, 0, stream>>>(x, wq, bq, wk, bk, wv, bv, Q, K, V);
    scores_kernel<<<dim3(32, 32, 4), 32, 0, stream>>>(st_emb, wst, mask, Q, K, scores);
    softmax_kernel<<<2048, 256, 0, stream>>>(scores);
    av_kernel<<<dim3(32, 8, 4), 32, 0, stream>>>(scores, V, attn_out);
    oproj_kernel<<<dim3(128, 8), 32, 0, stream>>>(attn_out, wo, bo, x, y);
    ln_kernel<<<256, 256, 0, stream>>>(y, lng, lnb, (float*)d_out);
}